// SPDVectorValuedDistance_69672959476270
// MI455X (gfx1250) — compile-verified
//
#include <hip/hip_runtime.h>
#include <math.h>

typedef float v2f __attribute__((ext_vector_type(2)));
typedef float v8f __attribute__((ext_vector_type(8)));

#define LDN 33            // padded LDS row stride (33 odd -> conflict-free column walks)
#define NS_ITERS 12       // Newton-Schulz iterations (first one folded into init)
#define JACOBI_SWEEPS 7   // cyclic Jacobi sweeps (quadratic convergence, n=32)

// D = A * B for 32x32 fp32 matrices resident in LDS (row-major, stride LDN).
// Uses V_WMMA_F32_16X16X4_F32: 2x2 tiles of 16x16, 8 k-chunks of 4 -> 32 WMMAs.
__device__ __forceinline__ void mm32(const float* A, const float* B, float* D,
                                     int lane) {
  const int half = lane >> 4;    // K-half select per ISA fragment layout
  const int lm   = lane & 15;
  v8f c00 = {}, c01 = {}, c10 = {}, c11 = {};
#pragma unroll
  for (int kc = 0; kc < 8; ++kc) {
    const int k0 = kc * 4 + half * 2;
    v2f a0, a1, b0, b1;
    // A-frag 16x4: lane l holds row (16I + l%16), K = k0, k0+1 (contiguous -> b64)
    a0.x = A[lm * LDN + k0];          a0.y = A[lm * LDN + k0 + 1];
    a1.x = A[(lm + 16) * LDN + k0];   a1.y = A[(lm + 16) * LDN + k0 + 1];
    // B-frag 4x16: VGPR v holds K=k0+v, N = 16J + l%16
    b0.x = B[k0 * LDN + lm];          b0.y = B[(k0 + 1) * LDN + lm];
    b1.x = B[k0 * LDN + lm + 16];     b1.y = B[(k0 + 1) * LDN + lm + 16];
    c00 = __builtin_amdgcn_wmma_f32_16x16x4_f32(false, a0, false, b0, (short)0, c00, false, false);
    c01 = __builtin_amdgcn_wmma_f32_16x16x4_f32(false, a0, false, b1, (short)0, c01, false, false);
    c10 = __builtin_amdgcn_wmma_f32_16x16x4_f32(false, a1, false, b0, (short)0, c10, false, false);
    c11 = __builtin_amdgcn_wmma_f32_16x16x4_f32(false, a1, false, b1, (short)0, c11, false, false);
  }
  // C/D layout: VGPR v -> rows v (lanes 0-15) and v+8 (lanes 16-31), col = lane%16
#pragma unroll
  for (int v = 0; v < 8; ++v) {
    D[(v + 8 * half) * LDN + lm]           = c00[v];
    D[(v + 8 * half) * LDN + lm + 16]      = c01[v];
    D[(16 + v + 8 * half) * LDN + lm]      = c10[v];
    D[(16 + v + 8 * half) * LDN + lm + 16] = c11[v];
  }
}

// round-robin tournament pairing: 31 rounds x 16 disjoint pairs covering all (i,j)
__device__ __forceinline__ void rr_pair(int r, int k, int& p, int& q) {
  if (k == 0) { p = 31; q = r; }
  else { p = (r + k) % 31; q = (r - k + 31) % 31; }
}

__global__ __launch_bounds__(32)
void spd_logeig_kernel(const float* __restrict__ x, const float* __restrict__ y,
                       float* __restrict__ out) {
  __shared__ float bufA[32 * LDN];
  __shared__ float bufX[32 * LDN];
  __shared__ float bufP[32 * LDN];
  __shared__ float bufQ[32 * LDN];
  __shared__ float csC[16], csS[16];

  const int lane = threadIdx.x;       // wave32: one wave == one block == one matrix
  const int b = blockIdx.x;
  const float* Ag = x + (size_t)b * 1024;
  const float* Bg = y + (size_t)b * 1024;

  // ---- load A (coalesced), Frobenius norm for normalization ----
  float ss = 0.f;
#pragma unroll
  for (int i = 0; i < 32; ++i) {
    float v = Ag[i * 32 + lane];
    bufA[i * LDN + lane] = v;
    ss += v * v;
  }
#pragma unroll
  for (int m = 16; m >= 1; m >>= 1) ss += __shfl_xor(ss, m, 32);
  const float fnorm = sqrtf(ss);        // >= lambda_max  -> A' eigvals in (0,1]
  const float invc = 1.0f / fnorm;
  __syncthreads();
  // A' = A/c ; X1 = 1.5I - 0.5A'  (Newton-Schulz step from X0 = I, done elementwise)
#pragma unroll
  for (int i = 0; i < 32; ++i) {
    float a = bufA[i * LDN + lane] * invc;
    bufA[i * LDN + lane] = a;
    bufX[i * LDN + lane] = (i == lane ? 1.5f : 0.0f) - 0.5f * a;
  }
  __syncthreads();

  float* pa = bufA; float* px = bufX; float* pp = bufP; float* pq = bufQ;
  // ---- Newton-Schulz: X <- X * (1.5I - 0.5 * A' * X * X)  -> A'^{-1/2} ----
  for (int it = 0; it < NS_ITERS - 1; ++it) {
    mm32(px, px, pp, lane);            // P = X*X
    __syncthreads();
    mm32(pa, pp, pq, lane);            // Q = A'*P
    __syncthreads();
#pragma unroll
    for (int i = 0; i < 32; ++i) {
      float qv = pq[i * LDN + lane];
      pq[i * LDN + lane] = (i == lane ? 1.5f : 0.0f) - 0.5f * qv;
    }
    __syncthreads();
    mm32(px, pq, pp, lane);            // Xnew = X*Q
    __syncthreads();
    float* t = px; px = pp; pp = t;
  }

  // ---- M = (X * B * X) / c  (A^{-1/2} = X / sqrt(c)) ----
#pragma unroll
  for (int i = 0; i < 32; ++i)
    pa[i * LDN + lane] = Bg[i * 32 + lane];   // A' dead; reuse for B
  __syncthreads();
  mm32(px, pa, pp, lane);              // P = X*B
  __syncthreads();
  mm32(pp, px, pq, lane);              // Q = P*X (unscaled M)
  __syncthreads();
  // symmetrize + scale into pp (read pq, write pp: no in-place hazard)
#pragma unroll
  for (int i = 0; i < 32; ++i) {
    float mij = pq[i * LDN + lane];
    float mji = pq[lane * LDN + i];
    pp[i * LDN + lane] = 0.5f * (mij + mji) * invc;
  }
  __syncthreads();
  float* M = pp;

  // ---- cyclic Jacobi eigenvalues ----
  for (int sw = 0; sw < JACOBI_SWEEPS; ++sw) {
    for (int r = 0; r < 31; ++r) {
      if (lane < 16) {                 // lane k computes angle for pair k
        int p, q; rr_pair(r, lane, p, q);
        float app = M[p * LDN + p];
        float aqq = M[q * LDN + q];
        float apq = M[p * LDN + q];
        float cc = 1.0f, s = 0.0f;
        if (fabsf(apq) > 1e-30f) {
          float tau = (aqq - app) / (2.0f * apq);
          float t = 1.0f / (fabsf(tau) + sqrtf(1.0f + tau * tau));
          t = (tau >= 0.0f) ? t : -t;
          cc = 1.0f / sqrtf(1.0f + t * t);
          s = t * cc;
        }
        csC[lane] = cc; csS[lane] = s;
      }
      __syncthreads();
      // row phase: M <- J^T M   (lane = column j; rows disjoint across pairs)
#pragma unroll 4
      for (int k = 0; k < 16; ++k) {
        int p, q; rr_pair(r, k, p, q);
        float cc = csC[k], s = csS[k];
        float mp = M[p * LDN + lane], mq = M[q * LDN + lane];
        M[p * LDN + lane] = cc * mp - s * mq;
        M[q * LDN + lane] = s * mp + cc * mq;
      }
      __syncthreads();
      // column phase: M <- M J   (lane = row i; padded stride -> no bank conflicts)
#pragma unroll 4
      for (int k = 0; k < 16; ++k) {
        int p, q; rr_pair(r, k, p, q);
        float cc = csC[k], s = csS[k];
        float mp = M[lane * LDN + p], mq = M[lane * LDN + q];
        M[lane * LDN + p] = cc * mp - s * mq;
        M[lane * LDN + q] = s * mp + cc * mq;
      }
      __syncthreads();
    }
  }

  // ---- sort eigenvalues ascending (bitonic across 32 lanes), log, store ----
  float d = M[lane * LDN + lane];
#pragma unroll
  for (int k = 2; k <= 32; k <<= 1) {
#pragma unroll
    for (int j = k >> 1; j > 0; j >>= 1) {
      float o = __shfl_xor(d, j, 32);
      bool dir = ((lane & k) == 0);
      bool low = ((lane & j) == 0);
      float mn = fminf(d, o), mx = fmaxf(d, o);
      d = (low == dir) ? mn : mx;
    }
  }
  out[(size_t)b * 32 + lane] = logf(d);
}

extern "C" void kernel_launch(void* const* d_in, const int* in_sizes, int n_in,
                              void* d_out, int out_size, void* d_ws, size_t ws_size,
                              hipStream_t stream) {
  const float* x = (const float*)d_in[0];
  const float* y = (const float*)d_in[1];
  float* out = (float*)d_out;
  const int batch = in_sizes[0] / 1024;   // 32x32 matrices
  spd_logeig_kernel<<<batch, 32, 0, stream>>>(x, y, out);
}